// FieldlinePusher_9122510537128
// MI455X (gfx1250) — compile-verified
//
#include <hip/hip_runtime.h>
#include <hip/hip_bf16.h>
#include <stdint.h>

#define BATCH   2048
#define NSTEPS  2048
#define NDENSE  256
#define NRES    128
#define TWO_PI  6.28318530717958647692f

struct F3 { float x, y, z; };
__device__ __forceinline__ F3 mk3(float a, float b, float c) { F3 r; r.x=a; r.y=b; r.z=c; return r; }
__device__ __forceinline__ F3 fma3(F3 a, float c, F3 b) { return mk3(a.x+c*b.x, a.y+c*b.y, a.z+c*b.z); }

struct Params { float raxis, zaxis, b0, q0, q2; };

// dy/dt for the field line (matches reference compute_bv / f)
__device__ __forceinline__ F3 fieldf(F3 yv, const Params& P) {
    float ct = __cosf(yv.z);
    float st = __sinf(yv.z);
    float r  = yv.x * ct + P.raxis;
    float z  = yv.x * st + P.zaxis;
    float dr = r - P.raxis;
    float dz = z - P.zaxis;
    float rho = sqrtf(dr*dr + dz*dz) + 1e-12f;
    float ctb = dr / rho;
    float stb = dz / rho;
    float bp  = P.b0 * rho / (P.raxis * (P.q0 + P.q2 * rho * rho));
    float br  = -bp * stb;
    float bt  = P.b0 * P.raxis / r;
    float bz  = bp * ctb;
    float drdt = br, dzdt = bz;
    return mk3(drdt*ct + dzdt*st, bt / r, (-drdt*st + dzdt*ct) / yv.x);
}

// One Dormand-Prince 5th-order step; k1 = f(y) supplied (FSAL-style reuse).
__device__ __forceinline__ F3 rkstep(F3 y, F3 k1, float dt, const Params& P) {
    F3 t  = fma3(y, dt*0.2f, k1);
    F3 k2 = fieldf(t, P);
    t = fma3(fma3(y, dt*(3.f/40.f), k1), dt*(9.f/40.f), k2);
    F3 k3 = fieldf(t, P);
    t = fma3(fma3(fma3(y, dt*(44.f/45.f), k1), dt*(-56.f/15.f), k2), dt*(32.f/9.f), k3);
    F3 k4 = fieldf(t, P);
    t = fma3(fma3(fma3(fma3(y, dt*(19372.f/6561.f), k1), dt*(-25360.f/2187.f), k2),
                  dt*(64448.f/6561.f), k3), dt*(-212.f/729.f), k4);
    F3 k5 = fieldf(t, P);
    t = fma3(fma3(fma3(fma3(fma3(y, dt*(9017.f/3168.f), k1), dt*(-355.f/33.f), k2),
                  dt*(46732.f/5247.f), k3), dt*(49.f/176.f), k4), dt*(-5103.f/18656.f), k5);
    F3 k6 = fieldf(t, P);
    F3 yn = fma3(fma3(fma3(fma3(fma3(y, dt*(35.f/384.f), k1), dt*(500.f/1113.f), k3),
                  dt*(125.f/192.f), k4), dt*(-2187.f/6784.f), k5), dt*(11.f/84.f), k6);
    return yn;
}

__device__ __forceinline__ float hermite(float s, float h, float y0, float y1, float f0, float f1) {
    float s2 = s*s, s3 = s2*s;
    return (2.f*s3 - 3.f*s2 + 1.f)*y0 + (s3 - 2.f*s2 + s)*h*f0
         + (-2.f*s3 + 3.f*s2)*y1 + (s3 - s2)*h*f1;
}
__device__ __forceinline__ float hermite_d(float s, float h, float y0, float y1, float f0, float f1) {
    float s2 = s*s;
    return (6.f*s2 - 6.f*s)*(y0 - y1) + (3.f*s2 - 4.f*s + 1.f)*h*f0 + (3.f*s2 - 2.f*s)*h*f1;
}

// ---------------------------------------------------------------------------
// Kernel A: integrate, find 2*pi crossing, Newton refine, write q; then
// re-integrate streaming the dense-output Hermite samples into d_ws.
// One particle per lane; 256 threads = 8 wave32 / block (2 waves per SIMD).
// ---------------------------------------------------------------------------
__global__ void __launch_bounds__(256)
kernel_integrate(const float* __restrict__ rhog0,
                 const float* __restrict__ p_raxis, const float* __restrict__ p_zaxis,
                 const float* __restrict__ p_b0,    const float* __restrict__ p_q0,
                 const float* __restrict__ p_q2,
                 float* __restrict__ out_q, float* __restrict__ ws_dense) {
    int p = blockIdx.x * blockDim.x + threadIdx.x;
    if (p >= BATCH) return;
    Params P; P.raxis = *p_raxis; P.zaxis = *p_zaxis; P.b0 = *p_b0; P.q0 = *p_q0; P.q2 = *p_q2;
    float max_t = 10.f * TWO_PI * P.raxis / (P.b0 * P.raxis);
    float dt = max_t / (float)NSTEPS;

    F3 y  = mk3(rhog0[p], 0.f, 0.f);
    F3 k1 = fieldf(y, P);
    F3 yp = y, fp = k1;                 // previous state/deriv
    F3 ys0 = y, fs0 = k1, ys1 = y, fs1 = k1;  // step-0/1 fallback (no crossing)
    bool found = false; int i0 = 0;
    F3 y0s = y, y1s = y, f0s = k1, f1s = k1;

    #pragma unroll 1
    for (int i = 0; i < NSTEPS; ++i) {
        if (!found && fabsf(y.z) >= TWO_PI) {   // first index with |theta|>=2pi
            found = true; i0 = i - 1;
            y0s = yp; f0s = fp; y1s = y; f1s = k1;
        }
        yp = y; fp = k1;
        y  = rkstep(y, k1, dt, P);
        k1 = fieldf(y, P);
        if (i == 0) { ys1 = y; fs1 = k1; }
    }
    if (!found && fabsf(y.z) >= TWO_PI) {       // crossing at index NSTEPS
        found = true; i0 = NSTEPS - 1;
        y0s = yp; f0s = fp; y1s = y; f1s = k1;
    }
    if (!found) {                                // argmax==0 path: i0 = clip(-1)=0
        i0 = 0; y0s = ys0; f0s = fs0; y1s = ys1; f1s = fs1;
    }

    float th0 = y0s.z, th1 = y1s.z;
    float target = (th1 > 0.f ? TWO_PI : (th1 < 0.f ? -TWO_PI : 0.f));
    float s = (target - th0) / (th1 - th0 + 1e-30f);
    #pragma unroll
    for (int it = 0; it < 3; ++it) {
        float g  = hermite(s, dt, th0, th1, f0s.z, f1s.z) - target;
        float gp = hermite_d(s, dt, th0, th1, f0s.z, f1s.z);
        s = fminf(fmaxf(s - g / (gp + 1e-30f), 0.f), 1.f);
    }
    float t_cross    = ((float)i0 + s) * dt;
    float varphi_end = hermite(s, dt, y0s.y, y1s.y, f0s.y, f1s.y);
    out_q[p] = varphi_end / TWO_PI;

    // Dense-output pass: query times are monotone, so stream during a second
    // integration instead of storing the 100 MB trajectory.
    y = mk3(rhog0[p], 0.f, 0.f);
    k1 = fieldf(y, P);
    float* th_buf  = ws_dense + (size_t)p * 512;   // [0..255] theta_q
    float* dth_buf = th_buf + 256;                 // [256..511] dtheta
    int qn = 0;
    #pragma unroll 1
    for (int i = 0; i < NSTEPS && qn < NDENSE; ++i) {
        F3 yn = rkstep(y, k1, dt, P);
        F3 kn = fieldf(yn, P);
        while (qn < NDENSE) {
            float tq = t_cross * ((float)qn * (1.f / (float)NDENSE));
            float u  = tq / dt;
            int iq = (int)floorf(u);
            iq = iq < 0 ? 0 : (iq > NSTEPS - 1 ? NSTEPS - 1 : iq);
            if (iq != i) break;                    // queries are monotone in i
            float sq = u - (float)iq;
            float vq = hermite(sq, dt, y.y, yn.y, k1.y, kn.y);
            float tg = hermite(sq, dt, y.z, yn.z, k1.z, kn.z);
            th_buf[qn]  = tg;
            dth_buf[qn] = tg - vq / varphi_end * TWO_PI;
            ++qn;
        }
        y = yn; k1 = kn;
    }
}

// ---------------------------------------------------------------------------
// Kernel B: resample dtheta(theta) onto uniform grid. One particle per block,
// one grid point per lane (128 threads = 4 wave32). The particle's 2 KB dense
// record is staged to LDS with the Tensor Data Mover (gfx1250 TDM), D# built
// per CDNA5 ISA section 8: 1-D tile of 512 x 4B elements.
// ---------------------------------------------------------------------------
typedef unsigned int v4u __attribute__((ext_vector_type(4)));
typedef int          v8i __attribute__((ext_vector_type(8)));
typedef int          v4i __attribute__((ext_vector_type(4)));

__global__ void __launch_bounds__(128)
kernel_resample(const float* __restrict__ ws_dense, float* __restrict__ out_dt) {
    __shared__ float buf[512];
    int p = blockIdx.x;
    int g = threadIdx.x;

    const float* src = ws_dense + (size_t)p * 512;
    unsigned long long ga = (unsigned long long)(uintptr_t)src;
    unsigned int lds_off  = (unsigned int)(uintptr_t)buf;   // low 32 bits = LDS byte offset

    // D# group 0: count=1 | lds_addr | global_addr[56:0] | type=2
    v4u d0;
    d0[0] = 1u;
    d0[1] = lds_off;
    d0[2] = (unsigned int)(ga & 0xFFFFFFFFu);
    d0[3] = ((unsigned int)((ga >> 32) & 0x01FFFFFFu)) | (2u << 30);
    // D# group 1: data_size=4B(code 2); tensor_dim0=512, tensor_dim1=1;
    // tile_dim0=512, tile_dim1=1; tensor_dim0_stride=512
    v8i d1;
    d1[0] = (int)(2u << 16);        // workgroup_mask=0, data_size=2 (4 bytes)
    d1[1] = (int)(512u << 16);      // tensor_dim0[15:0]
    d1[2] = (int)(1u << 16);        // tensor_dim0[31:16]=0 | tensor_dim1[15:0]=1
    d1[3] = (int)(512u << 16);      // tensor_dim1[31:16]=0 | tile_dim0=512
    d1[4] = 1;                      // tile_dim1=1 | tile_dim2=0
    d1[5] = 512;                    // tensor_dim0_stride[31:0]
    d1[6] = 0;
    d1[7] = 0;
    v4i d2 = {0, 0, 0, 0};
    v4i d3 = {0, 0, 0, 0};
    v8i dx = {0, 0, 0, 0, 0, 0, 0, 0};   // extra group (clang-23 6-arg form); zeros
    __builtin_amdgcn_tensor_load_to_lds(d0, d1, d2, d3, dx, 0);
    __builtin_amdgcn_s_wait_tensorcnt((short)0);
    __syncthreads();

    const float* th = buf;        // theta_q[0..255]
    const float* dh = buf + 256;  // dtheta[0..255]

    // Extended arrays (length 260), built on the fly from LDS:
    auto XE = [&](int k) -> float {
        return (k < 2) ? th[254 + k] - TWO_PI : ((k < 258) ? th[k - 2] : th[k - 258] + TWO_PI);
    };
    auto YE = [&](int k) -> float {
        return (k < 2) ? dh[254 + k] : ((k < 258) ? dh[k - 2] : dh[k - 258]);
    };

    float v = (float)g * (TWO_PI / (float)NRES);
    // searchsorted(xe, v, side='left'): first idx with xe[idx] >= v
    int lo = 0, hi = 260;
    while (lo < hi) {
        int mid = (lo + hi) >> 1;
        if (XE(mid) < v) lo = mid + 1; else hi = mid;
    }
    int j = lo - 1;
    j = j < 1 ? 1 : (j > 257 ? 257 : j);

    float x0 = XE(j),   x1 = XE(j + 1);
    float yy0 = YE(j),  yy1 = YE(j + 1);
    float m0 = (YE(j + 1) - YE(j - 1)) / (XE(j + 1) - XE(j - 1) + 1e-30f);  // m[j-1]
    float m1 = (YE(j + 2) - YE(j))     / (XE(j + 2) - XE(j)     + 1e-30f);  // m[j]
    float h  = x1 - x0;
    float uu = (v - x0) / (h + 1e-30f);
    out_dt[(size_t)p * NRES + g] = hermite(uu, h, yy0, yy1, m0, m1);
}

extern "C" void kernel_launch(void* const* d_in, const int* in_sizes, int n_in,
                              void* d_out, int out_size, void* d_ws, size_t ws_size,
                              hipStream_t stream) {
    const float* rhog0 = (const float*)d_in[0];
    const float* raxis = (const float*)d_in[1];
    const float* zaxis = (const float*)d_in[2];
    const float* b0    = (const float*)d_in[3];
    const float* q0    = (const float*)d_in[4];
    const float* q2    = (const float*)d_in[5];
    float* out = (float*)d_out;               // [0..B) = q, [B..B+B*NRES) = dtheta_resamp
    float* ws  = (float*)d_ws;                // B*512 floats of dense-output records (4 MB)

    kernel_integrate<<<(BATCH + 255) / 256, 256, 0, stream>>>(
        rhog0, raxis, zaxis, b0, q0, q2, out, ws);
    kernel_resample<<<BATCH, NRES, 0, stream>>>(ws, out + BATCH);
}